// SparseLRRLayer_86088324481668
// MI455X (gfx1250) — compile-verified
//
#include <hip/hip_runtime.h>
#include <hip/hip_bf16.h>

#define D_FEAT 128
#define LAMBDA_REG 1.0f
#define GAMMA 0.1f
#define EPS_F 1e-8f

typedef float v2f __attribute__((ext_vector_type(2)));
typedef float v8f __attribute__((ext_vector_type(8)));

// Exact f32 wave32 sum. One V_WMMA_F32_16X16X4_F32 with A = ones(16x4) computes
// column sums of B: D[0][n] = B[0][n] + B[1][n] = x[lane n] + x[lane n+16]
// (B rows K=0/K=1 are lanes 0-15 / 16-31 of b[0]; K=2,3 are zeroed).
// All math is exact f32, so correctness is preserved; remaining 16-lane fold via shfl.
__device__ __forceinline__ float waveReduceSum(float x) {
#if __has_builtin(__builtin_amdgcn_wmma_f32_16x16x4_f32)
  v2f a; a[0] = 1.0f; a[1] = 1.0f;   // A-matrix: all ones (16x4)
  v2f b; b[0] = x;    b[1] = 0.0f;   // B-matrix rows 0,1 = lane values; rows 2,3 = 0
  v8f c = {};
  c = __builtin_amdgcn_wmma_f32_16x16x4_f32(false, a, false, b, (short)0, c, false, false);
  float s = c[0];                    // lane L and L^16 hold x[L&15] + x[(L&15)+16]
  s += __shfl_xor(s, 8, 32);
  s += __shfl_xor(s, 4, 32);
  s += __shfl_xor(s, 2, 32);
  s += __shfl_xor(s, 1, 32);
  return s;
#else
  for (int m = 16; m >= 1; m >>= 1) x += __shfl_xor(x, m, 32);
  return x;
#endif
}

__device__ __forceinline__ void atomicAddF(float* p, float v) {
  unsafeAtomicAdd(p, v);             // native global_atomic_add_f32, no CAS loop
}

// ---------------- Pass 1: column sums (scatter-add) ----------------
__global__ void k_colsum(const float* __restrict__ C, const int* __restrict__ col,
                         float* __restrict__ colS, int E) {
  int e = blockIdx.x * blockDim.x + threadIdx.x;
  if (e < E) atomicAddF(&colS[col[e]], C[e]);
}

// ---------------- Pass 2: norm_vals, reg partial, |v| top-16 histogram ----------------
__global__ void k_norm(const float* __restrict__ C, const int* __restrict__ col,
                       const float* __restrict__ colS, float* __restrict__ normv,
                       unsigned* __restrict__ hist1, float* __restrict__ scal, int E) {
  int e = blockIdx.x * blockDim.x + threadIdx.x;
  float part = 0.f;
  if (e < E) {
    float cs = colS[col[e]] + EPS_F;
    float v = C[e] / cs;
    normv[e] = v;
    part = v * v;
    unsigned bits = __float_as_uint(fabsf(v));
    atomicAdd(&hist1[bits >> 16], 1u);
  }
  float s = waveReduceSum(part);     // uniform point, EXEC all-1s
  if ((threadIdx.x & 31) == 0) atomicAddF(&scal[1], s);
}

// ---------------- Pass 3: Z_recon += v * Z[row] (one wave per edge) ----------------
__global__ void k_scatter(const float* __restrict__ Z, const float* __restrict__ normv,
                          const int* __restrict__ row, const int* __restrict__ col,
                          float* __restrict__ Zrec, int E) {
  int wave = (int)((blockIdx.x * (unsigned)blockDim.x + threadIdx.x) >> 5);
  int lane = threadIdx.x & 31;
  if (wave >= E) return;
  int r = row[wave];
  int c = col[wave];
  float v = normv[wave];
  // 32 lanes x float4 = 512B fully coalesced row gather
  float4 z = reinterpret_cast<const float4*>(Z + (size_t)r * D_FEAT)[lane];
  float* dst = Zrec + (size_t)c * D_FEAT + lane * 4;
  atomicAddF(dst + 0, v * z.x);
  atomicAddF(dst + 1, v * z.y);
  atomicAddF(dst + 2, v * z.z);
  atomicAddF(dst + 3, v * z.w);
}

// ---------------- Radix-select helper: scan 65536-bin histogram, find bin of `rank` ----------------
__device__ void scanFind(const unsigned* __restrict__ hist, unsigned rank,
                         unsigned* __restrict__ outBinCum, unsigned* sh) {
  int t = threadIdx.x;               // 256 threads, 256 bins each
  unsigned s = 0;
  for (int i = 0; i < 256; ++i) s += hist[t * 256 + i];
  sh[t] = s; __syncthreads();
  for (int off = 1; off < 256; off <<= 1) {   // inclusive Hillis-Steele scan
    unsigned mine = sh[t];
    unsigned add = (t >= off) ? sh[t - off] : 0u;
    __syncthreads();
    sh[t] = mine + add;
    __syncthreads();
  }
  unsigned cum = sh[t] - s;          // exclusive prefix of this thread's chunk
  for (int i = 0; i < 256; ++i) {
    unsigned c = hist[t * 256 + i];
    if (cum <= rank && rank < cum + c) {       // unique thread writes
      outBinCum[0] = (unsigned)(t * 256 + i);
      outBinCum[1] = cum;
    }
    cum += c;
  }
  __syncthreads();
}

__global__ void k_select1(const unsigned* __restrict__ hist, unsigned* __restrict__ sel,
                          unsigned k1, unsigned k2) {
  __shared__ unsigned sh[256];
  scanFind(hist, k1, &sel[0], sh);   // sel[0]=bin1, sel[1]=count-before
  scanFind(hist, k2, &sel[2], sh);   // sel[2]=bin2, sel[3]=count-before
}

// ---------------- Refine: low-16-bit histograms inside the two candidate bins ----------------
__global__ void k_hist2(const float* __restrict__ normv, const unsigned* __restrict__ sel,
                        unsigned* __restrict__ h2a, unsigned* __restrict__ h2b, int E) {
  int e = blockIdx.x * blockDim.x + threadIdx.x;
  if (e >= E) return;
  unsigned bits = __float_as_uint(fabsf(normv[e]));
  unsigned top = bits >> 16, low = bits & 0xFFFFu;
  if (top == sel[0]) atomicAdd(&h2a[low], 1u);
  if (top == sel[2]) atomicAdd(&h2b[low], 1u);
}

__global__ void k_select2(const unsigned* __restrict__ h2a, const unsigned* __restrict__ h2b,
                          unsigned* __restrict__ sel, float* __restrict__ scal,
                          unsigned k1, unsigned k2) {
  __shared__ unsigned sh[256];
  unsigned b1 = sel[0], c1 = sel[1], b2 = sel[2], c2 = sel[3];
  scanFind(h2a, k1 - c1, &sel[4], sh);
  scanFind(h2b, k2 - c2, &sel[6], sh);
  __syncthreads();
  if (threadIdx.x == 0) {
    float v1 = __uint_as_float((b1 << 16) | (sel[4] & 0xFFFFu));
    float v2 = __uint_as_float((b2 << 16) | (sel[6] & 0xFFFFu));
    scal[3] = 0.5f * (v1 + v2);      // median of |norm_vals| (even-count average)
  }
}

// ---------------- Masked block-loss sum ----------------
__global__ void k_blockloss(const float* __restrict__ normv, float* __restrict__ scal, int E) {
  int e = blockIdx.x * blockDim.x + threadIdx.x;
  float th = scal[3];
  float part = 0.f;
  if (e < E) {
    float v = normv[e];
    if (fabsf(v) < th) part = v * v;
  }
  float s = waveReduceSum(part);
  if ((threadIdx.x & 31) == 0) atomicAddF(&scal[2], s);
}

// ---------------- Reconstruction loss sum ----------------
__global__ void k_recon(const float* __restrict__ Z, const float* __restrict__ Zrec,
                        float* __restrict__ scal, long long n4) {
  long long i = (long long)blockIdx.x * blockDim.x + threadIdx.x;
  float part = 0.f;
  if (i < n4) {
    float4 a = reinterpret_cast<const float4*>(Zrec)[i];
    float4 b = reinterpret_cast<const float4*>(Z)[i];
    float dx = a.x - b.x, dy = a.y - b.y, dz = a.z - b.z, dw = a.w - b.w;
    part = dx * dx + dy * dy + dz * dz + dw * dw;
  }
  float s = waveReduceSum(part);
  if ((threadIdx.x & 31) == 0) atomicAddF(&scal[0], s);
}

__global__ void k_final(const float* __restrict__ scal, float* __restrict__ out3, float invCount) {
  out3[0] = scal[0] * invCount;      // recon_loss = mean
  out3[1] = LAMBDA_REG * scal[1];    // reg_loss
  out3[2] = GAMMA * scal[2];         // block_loss
}

extern "C" void kernel_launch(void* const* d_in, const int* in_sizes, int n_in,
                              void* d_out, int out_size, void* d_ws, size_t ws_size,
                              hipStream_t stream) {
  (void)n_in; (void)out_size; (void)ws_size;
  const float* Z   = (const float*)d_in[0];
  const float* C   = (const float*)d_in[1];
  const int*   row = (const int*)d_in[2];
  const int*   col = (const int*)d_in[3];
  const int E = in_sizes[1];
  const int N = in_sizes[0] / D_FEAT;
  float* out = (float*)d_out;

  // Workspace layout
  float*    Zrec  = (float*)d_ws;                     // N*128 f32 (51.2 MB, L2-resident)
  float*    colS  = Zrec + (size_t)N * D_FEAT;        // N f32
  unsigned* hist1 = (unsigned*)(colS + N);            // 65536
  unsigned* h2a   = hist1 + 65536;                    // 65536
  unsigned* h2b   = h2a + 65536;                      // 65536
  unsigned* sel   = h2b + 65536;                      // 8
  float*    scal  = (float*)(sel + 8);                // [recon, reg, block, threshold]
  size_t zero_bytes = ((size_t)N * D_FEAT + N + 3u * 65536u + 8u + 8u) * 4u;
  hipMemsetAsync(d_ws, 0, zero_bytes, stream);        // fresh accumulators every call

  const int tb = 256;
  const int gE = (E + tb - 1) / tb;

  k_colsum<<<gE, tb, 0, stream>>>(C, col, colS, E);
  k_norm<<<gE, tb, 0, stream>>>(C, col, colS, out, hist1, scal, E);

  const int gScatter = (int)(((long long)E * 32 + tb - 1) / tb);
  k_scatter<<<gScatter, tb, 0, stream>>>(Z, out, row, col, Zrec, E);

  unsigned k1, k2;
  if (E & 1) { k1 = k2 = (unsigned)(E / 2); }
  else       { k1 = (unsigned)(E / 2 - 1); k2 = (unsigned)(E / 2); }
  k_select1<<<1, 256, 0, stream>>>(hist1, sel, k1, k2);
  k_hist2<<<gE, tb, 0, stream>>>(out, sel, h2a, h2b, E);
  k_select2<<<1, 256, 0, stream>>>(h2a, h2b, sel, scal, k1, k2);

  k_blockloss<<<gE, tb, 0, stream>>>(out, scal, E);

  const long long n4 = (long long)N * D_FEAT / 4;
  const int gR = (int)((n4 + tb - 1) / tb);
  k_recon<<<gR, tb, 0, stream>>>(Z, Zrec, scal, n4);

  k_final<<<1, 1, 0, stream>>>(scal, out + E, 1.0f / ((float)N * (float)D_FEAT));
}